// TopKBagModel_970662609155
// MI455X (gfx1250) — compile-verified
//
#include <hip/hip_runtime.h>
#include <stdint.h>

#define N_COLS   8192
#define NTHREADS 512
#define NCHUNK4  (N_COLS / 4 / NTHREADS)   // 4 uint4-chunks per thread
#define CAND_MAX 2048                      // >= max k (1638) + float-tie headroom

typedef uint32_t u32;
typedef uint64_t u64;
typedef __attribute__((ext_vector_type(4))) unsigned int v4u;
typedef __attribute__((ext_vector_type(8))) int          v8i;
typedef __attribute__((ext_vector_type(4))) int          v4i;

__global__ __launch_bounds__(NTHREADS)
void topk_bag_kernel(const float* __restrict__ X, const int* __restrict__ mask,
                     float* __restrict__ bag, int* __restrict__ keys) {
    __shared__ u32 kb[N_COLS];        // staged X bits -> masked value bits
    __shared__ u64 cand[CAND_MAX];    // composite keys: (valbits<<32)|(N-1-idx)
    __shared__ u32 hist[256];
    __shared__ u32 scan[256];         // suffix sums of hist
    __shared__ u32 s_prefix, s_t, s_k, s_nvalid, s_count;

    const int row = blockIdx.x;
    const int tid = threadIdx.x;

    // ---- Phase 0: TDM async DMA of the X row (32 KB) into LDS ----
    if (tid == 0) {
        u64 gx   = (u64)(uintptr_t)(X + (size_t)row * N_COLS);
        u32 ldsx = (u32)(uintptr_t)&kb[0];
        // D# group0: count=1, lds_addr, global_addr[56:0], type=2
        v4u g0 = { 1u, ldsx, (u32)gx, (u32)((gx >> 32) & 0x01FFFFFFu) | 0x80000000u };
        // group1: data_size=4B; tensor_dim0=8192; tensor_dim1=1; tile_dim0=8192;
        //         tile_dim1=1; tensor_dim0_stride=8192
        v8i g1 = { 0x00020000, 0x20000000, 0x00010000, 0x20000000, 1, 8192, 0, 0 };
        v4i z4 = { 0, 0, 0, 0 };
#if __clang_major__ >= 23
        v8i z8 = { 0, 0, 0, 0, 0, 0, 0, 0 };
        __builtin_amdgcn_tensor_load_to_lds(g0, g1, z4, z4, z8, 0);
#else
        __builtin_amdgcn_tensor_load_to_lds(g0, g1, z4, z4, 0);
#endif
    }

    // Mask row via b128 vector loads, overlapping the TDM transfer.
    const uint4* mrow4 = (const uint4*)(mask + (size_t)row * N_COLS);
    uint4 mv[NCHUNK4];
#pragma unroll
    for (int ch = 0; ch < NCHUNK4; ++ch)
        mv[ch] = mrow4[tid + ch * NTHREADS];

    if (tid == 0) {
        __builtin_amdgcn_s_wait_tensorcnt(0);   // TDM complete -> kb[] valid
        s_nvalid = 0u; s_count = 0u;
    }
    __syncthreads();

    // ---- Phase A: masked key bits (b128 LDS rd/wr) + valid count ----
    uint4* kb4 = (uint4*)kb;
    u32 lv = 0;
#pragma unroll
    for (int ch = 0; ch < NCHUNK4; ++ch) {
        int c4 = tid + ch * NTHREADS;
        uint4 xv = kb4[c4];
        uint4 m  = mv[ch];
        xv.x = (m.x & 1u) ? xv.x : 0u;  lv += (m.x & 1u);
        xv.y = (m.y & 1u) ? xv.y : 0u;  lv += (m.y & 1u);
        xv.z = (m.z & 1u) ? xv.z : 0u;  lv += (m.z & 1u);
        xv.w = (m.w & 1u) ? xv.w : 0u;  lv += (m.w & 1u);
        kb4[c4] = xv;
    }
    atomicAdd(&s_nvalid, lv);
    __syncthreads();

    if (tid == 0) {
        int nv = (int)s_nvalid;
        int kk = (int)floorf(0.2f * (float)nv);
        if (kk < 1) kk = 1;
        s_k = (u32)kk; s_prefix = 0u; s_t = (u32)kk;
    }
    __syncthreads();

    // ---- Phase B: 4-pass MSD radix select of the k-th largest value bits ----
    for (int pass = 0; pass < 4; ++pass) {
        const int shift = 24 - 8 * pass;
        if (tid < 256) hist[tid] = 0u;
        __syncthreads();
        const u32 pfx = s_prefix;
        const u32 t   = s_t;
#pragma unroll
        for (int ch = 0; ch < NCHUNK4; ++ch) {
            uint4 v = kb4[tid + ch * NTHREADS];
            u32 e0 = v.x, e1 = v.y, e2 = v.z, e3 = v.w;
            if (pass == 0 || (e0 >> (shift + 8)) == pfx) atomicAdd(&hist[(e0 >> shift) & 0xFFu], 1u);
            if (pass == 0 || (e1 >> (shift + 8)) == pfx) atomicAdd(&hist[(e1 >> shift) & 0xFFu], 1u);
            if (pass == 0 || (e2 >> (shift + 8)) == pfx) atomicAdd(&hist[(e2 >> shift) & 0xFFu], 1u);
            if (pass == 0 || (e3 >> (shift + 8)) == pfx) atomicAdd(&hist[(e3 >> shift) & 0xFFu], 1u);
        }
        __syncthreads();
        // Parallel suffix scan: scan[d] = sum_{d'>=d} hist[d']  (8 Hillis-Steele steps)
        if (tid < 256) scan[tid] = hist[tid];
        __syncthreads();
        for (int off = 1; off < 256; off <<= 1) {
            u32 v = 0;
            if (tid < 256) {
                v = scan[tid];
                if (tid + off < 256) v += scan[tid + off];
            }
            __syncthreads();
            if (tid < 256) scan[tid] = v;
            __syncthreads();
        }
        // Winning bin d: count(> d) < t <= count(>= d); unique thread writes.
        if (tid < 256) {
            u32 gt = (tid < 255) ? scan[tid + 1] : 0u;
            if (gt < t && scan[tid] >= t) {
                s_prefix = (pfx << 8) | (u32)tid;
                s_t      = t - gt;
            }
        }
        __syncthreads();
    }
    const u32 vk = s_prefix;          // exact k-th largest masked value (bit pattern)
    const u32 k  = s_k;

    // ---- Phase C: compact nonzero candidates >= vk with composite keys ----
#pragma unroll
    for (int ch = 0; ch < NCHUNK4; ++ch) {
        int base = (tid + ch * NTHREADS) * 4;
        uint4 v = kb4[tid + ch * NTHREADS];
        u32 e[4] = { v.x, v.y, v.z, v.w };
#pragma unroll
        for (int l = 0; l < 4; ++l) {
            if (e[l] >= vk && e[l] != 0u) {
                u32 p = atomicAdd(&s_count, 1u);
                if (p < CAND_MAX)
                    cand[p] = ((u64)e[l] << 32) | (u32)(N_COLS - 1 - (base + l));
            }
        }
    }
    __syncthreads();
    u32 c = s_count; if (c > CAND_MAX) c = CAND_MAX;
    u32 M = 1; while (M < c) M <<= 1;                 // pow2 size for bitonic
    for (u32 j = c + tid; j < M; j += NTHREADS) cand[j] = 0ull;   // pads sort last
    __syncthreads();

    // ---- Phase D: bitonic sort descending (value desc, index asc via key) ----
    for (u32 size = 2; size <= M; size <<= 1) {
        for (u32 stride = size >> 1; stride > 0; stride >>= 1) {
            for (u32 t2 = tid; t2 < (M >> 1); t2 += NTHREADS) {
                u32 i = ((t2 & ~(stride - 1)) << 1) | (t2 & (stride - 1));
                u32 j = i | stride;
                u64 a = cand[i], b = cand[j];
                bool desc = ((i & size) == 0);
                if ((a < b) == desc) { cand[i] = b; cand[j] = a; }
            }
            __syncthreads();
        }
    }

    // Rare path: fewer than k nonzero candidates -> smallest-index zeros fill
    if (tid == 0 && c < k) {
        u32 need = k - c, got = 0;
        for (int i = 0; i < N_COLS && got < need; ++i)
            if (kb[i] == 0u) { cand[c + got] = (u64)(u32)(N_COLS - 1 - i); ++got; }
    }
    __syncthreads();

    // ---- Phase E: outputs. Key indices via b128 stores (dominant HBM traffic) ----
    int4* krow4 = (int4*)(keys + (size_t)row * N_COLS);
#pragma unroll
    for (int ch = 0; ch < NCHUNK4; ++ch) {
        int c4 = tid + ch * NTHREADS;
        int j  = c4 * 4;
        int4 out = { -100, -100, -100, -100 };
        if ((u32)j < k)       out.x = (int)(N_COLS - 1 - (u32)(cand[j + 0] & 0xFFFFFFFFu));
        if ((u32)(j + 1) < k) out.y = (int)(N_COLS - 1 - (u32)(cand[j + 1] & 0xFFFFFFFFu));
        if ((u32)(j + 2) < k) out.z = (int)(N_COLS - 1 - (u32)(cand[j + 2] & 0xFFFFFFFFu));
        if ((u32)(j + 3) < k) out.w = (int)(N_COLS - 1 - (u32)(cand[j + 3] & 0xFFFFFFFFu));
        krow4[c4] = out;
    }
    if (tid == 0) {
        // Sequential descending-order sum == reference cumsum order (bit-exact,
        // deterministic); latency hidden by co-resident workgroups.
        float s = 0.0f;
        for (u32 j = 0; j < k; ++j) s += __uint_as_float((u32)(cand[j] >> 32));
        bag[row] = s / (float)k;
    }
}

extern "C" void kernel_launch(void* const* d_in, const int* in_sizes, int n_in,
                              void* d_out, int out_size, void* d_ws, size_t ws_size,
                              hipStream_t stream) {
    const float* X    = (const float*)d_in[0];
    const int*   mask = (const int*)d_in[1];
    const int Brows = in_sizes[0] / N_COLS;
    float* bag  = (float*)d_out;            // [B] bag probabilities (float32)
    int*   keys = (int*)(bag + Brows);      // [B, N] key-instance indices (int32)
    hipLaunchKernelGGL(topk_bag_kernel, dim3(Brows), dim3(NTHREADS), 0, stream,
                       X, mask, bag, keys);
}